// CollapseAwareEmbedding_63118839382531
// MI455X (gfx1250) — compile-verified
//
#include <hip/hip_runtime.h>
#include <math.h>

// ---------------------------------------------------------------------------
// CollapseAwareEmbedding for MI455X (gfx1250, wave32)
//   out = [ s_out (1096*384 f32) | z (1096*1096*128 f32, ~615 MB) ]
// z is pure streaming-store bound (~26 us @ 23.3 TB/s); s_out uses
// V_WMMA_F32_16X16X4_F32 for the 1095x22 @ 22x384 GEMM (K padded to 24).
// ---------------------------------------------------------------------------

typedef __attribute__((ext_vector_type(2))) float v2f;
typedef __attribute__((ext_vector_type(4))) float v4f;
typedef __attribute__((ext_vector_type(8))) float v8f;

#define S_DIM   384
#define S_IN    21
#define L_TOT   1096          // 1 collapse + 1095 residue rows
#define SROWS   1095
#define MTILES  69            // ceil(1095/16)
#define NTILES  24            // 384/16
#define TILES_TOT (MTILES * NTILES)   // 1656 = 207 * 8
#define WPB     8             // waves per 256-thread block
#define ZPAIRS  (L_TOT * L_TOT)
#define S_ELEMS (L_TOT * S_DIM)

// region of a non-collapse s-row r in [0,1094]; k=-1 is "hd", k=0..5 conds
__device__ __forceinline__ void sreg(int r, int& k, int& local) {
  if      (r < 400)  { k = -1; local = r; }
  else if (r < 800)  { k = 0;  local = r - 400; }
  else if (r < 815)  { k = 1;  local = r - 800; }
  else if (r < 935)  { k = 2;  local = r - 815; }
  else if (r < 955)  { k = 3;  local = r - 935; }
  else if (r < 1075) { k = 4;  local = r - 955; }
  else               { k = 5;  local = r - 1075; }
}

// region id for the 1096-long pair axis (0 = collapse token)
__device__ __forceinline__ int rid_of(int t) {
  if (t == 0)    return 0;
  int r = t - 1;
  if (r < 400)   return 1;
  if (r < 800)   return 2;
  if (r < 815)   return 3;
  if (r < 935)   return 4;
  if (r < 955)   return 5;
  if (r < 1075)  return 6;
  return 7;
}

__device__ __forceinline__ int pid_of(int i, int j) {
  int ri = rid_of(i), rj = rid_of(j);
  if (ri == 0 || rj == 0) return (ri == 0 && rj == 0) ? 0 : 1;
  if (ri == 1 && rj == 1) {
    int d = (i > j) ? (i - j) : (j - i);
    if (d == 1) return 2;
    if (d > 1)  return 3;
    return 0;                       // diagonal stays 0 (np.zeros default)
  }
  if (ri == 1 || rj == 1) return 4; // cross hd<->cond
  if (ri == rj) return 5 + (ri - 2);
  int a = ((ri < rj) ? ri : rj) - 2;
  int b = ((ri > rj) ? ri : rj) - 2;
  int p = 11 + a * 5 - (a * (a - 1)) / 2 + (b - a - 1);  // n_cond = 6
  return p > 31 ? 31 : p;
}

// padded A column: col 0 = mask (hd) or 0, cols 1..21 = x, cols 22..23 = 0
__device__ __forceinline__ float aacol(const float* xrow, float a0, int c) {
  if (c == 0)  return a0;
  if (c <= 21) return xrow[c - 1];
  return 0.0f;
}

// ---------------------------------------------------------------------------
// Kernel 1: s_out rows 1..1095 via V_WMMA_F32_16X16X4_F32 + pos epilogue
// ---------------------------------------------------------------------------
__global__ __launch_bounds__(256)
void emb_kernel(const float* __restrict__ hd,  const float* __restrict__ mask,
                const float* __restrict__ mhc, const float* __restrict__ pep,
                const float* __restrict__ lv,  const float* __restrict__ lj,
                const float* __restrict__ hv,  const float* __restrict__ hj,
                const int* __restrict__ mhc_idx, const int* __restrict__ pep_idx,
                const int* __restrict__ lv_idx,  const int* __restrict__ lj_idx,
                const int* __restrict__ hv_idx,  const int* __restrict__ hj_idx,
                const float* __restrict__ seq_W, const float* __restrict__ seq_b,
                const float* __restrict__ pos_W, const float* __restrict__ pos_b,
                const float* __restrict__ region_w, float* __restrict__ out)
{
  // freq[K] = pi / 2056^(K/32)
  __shared__ float freq[32];
  if (threadIdx.x < 32)
    freq[threadIdx.x] =
        3.14159265358979f * __expf(-(float)threadIdx.x * (7.6285188f / 32.0f));

  const int lane  = threadIdx.x & 31;
  const int wv    = threadIdx.x >> 5;
  const int tile  = blockIdx.x * WPB + wv;        // grid is exact: 207*8=1656
  const int mt    = tile / NTILES;
  const int nt    = tile - mt * NTILES;

  const int lm    = lane & 15;                    // M (for A) / N (for B,C)
  const int khalf = (lane >> 4) << 1;             // K sub-block: 0 or 2
  const int ncol  = nt * 16 + lm;

  // ---- A-fragment row setup (row r feeds only C row r => clamp is safe) ----
  int r  = mt * 16 + lm;
  int rr = (r < SROWS) ? r : (SROWS - 1);
  int k, local;
  sreg(rr, k, local);
  const float* xrow;
  switch (k) {
    case -1: xrow = hd  + local * S_IN; break;
    case 0:  xrow = mhc + local * S_IN; break;
    case 1:  xrow = pep + local * S_IN; break;
    case 2:  xrow = lv  + local * S_IN; break;
    case 3:  xrow = lj  + local * S_IN; break;
    case 4:  xrow = hv  + local * S_IN; break;
    default: xrow = hj  + local * S_IN; break;
  }
  const float a0 = (k == -1) ? mask[local] : 0.0f;

  // ---- GEMM: 6 x V_WMMA_F32_16X16X4_F32 (K = 24, last 2 cols zero-pad) ----
  v8f acc = {};
#pragma unroll
  for (int kk = 0; kk < 24; kk += 4) {
    const int c0 = kk + khalf;
    v2f a, b;
    a.x = aacol(xrow, a0, c0);
    a.y = aacol(xrow, a0, c0 + 1);
    b.x = (c0     < 22) ? seq_W[c0 * S_DIM + ncol]       : 0.0f;
    b.y = (c0 + 1 < 22) ? seq_W[(c0 + 1) * S_DIM + ncol] : 0.0f;
    acc = __builtin_amdgcn_wmma_f32_16x16x4_f32(
        /*neg_a=*/false, a, /*neg_b=*/false, b,
        /*c_mod=*/(short)0, acc, /*reuse_a=*/false, /*reuse_b=*/false);
  }

  __syncthreads();   // freq[] ready for epilogue

  // ---- epilogue: +seq_b, optional positional blend, store ----
  const float sb = seq_b[ncol];
#pragma unroll
  for (int v = 0; v < 8; ++v) {
    const int g = mt * 16 + v + ((lane >> 4) << 3);   // C row of this VGPR
    if (g >= SROWS) continue;
    const float seqv = acc[v] + sb;
    int k2, loc2;
    sreg(g, k2, loc2);
    float outv;
    if (k2 < 0) {
      outv = seqv;                                    // hd block: no pos term
    } else {
      const int* idxp;
      switch (k2) {
        case 0:  idxp = mhc_idx; break;
        case 1:  idxp = pep_idx; break;
        case 2:  idxp = lv_idx;  break;
        case 3:  idxp = lj_idx;  break;
        case 4:  idxp = hv_idx;  break;
        default: idxp = hj_idx;  break;
      }
      const float fi = (float)idxp[loc2];
      float pos = pos_b[ncol];
#pragma unroll 4
      for (int K = 0; K < 32; ++K) {
        float s, c;
        __sincosf(fi * freq[K], &s, &c);
        pos = __fmaf_rn(s, pos_W[K * S_DIM + ncol], pos);
        pos = __fmaf_rn(c, pos_W[(K + 32) * S_DIM + ncol], pos);
      }
      outv = region_w[2 * k2] * seqv + region_w[2 * k2 + 1] * pos;
    }
    out[(size_t)(g + 1) * S_DIM + ncol] = outv;
  }
}

// ---------------------------------------------------------------------------
// Kernel 2: z — one wave32 per (i,j) pair; LDS 32x128 table; NT b128 stores
// ---------------------------------------------------------------------------
__global__ __launch_bounds__(256)
void z_kernel(const float* __restrict__ tab1, const float* __restrict__ b1,
              const float* __restrict__ tab2, const float* __restrict__ b2,
              float* __restrict__ out)
{
  __shared__ __align__(16) float ztab[32 * 128];    // 16 KB: all 32 pid rows
  for (int t = threadIdx.x; t < 32 * 128; t += 256) {
    const int pid = t >> 7;
    const int c   = t & 127;
    ztab[t] = (c < 64) ? (tab1[(pid >> 2) * 64 + c] + b1[c])
                       : (tab2[(pid & 3) * 64 + (c - 64)] + b2[c - 64]);
  }
  __syncthreads();

  const int lane = threadIdx.x & 31;
  const int wv   = threadIdx.x >> 5;
  const unsigned int p = blockIdx.x * (unsigned)WPB + (unsigned)wv;
  if (p < (unsigned)ZPAIRS) {
    const int i   = (int)(p / 1096u);
    const int j   = (int)(p - (unsigned)i * 1096u);
    const int pid = pid_of(i, j);
    // ds_load_b128 from the pid row, global_store_b128 non-temporal
    const v4f val = *(const v4f*)(ztab + pid * 128 + lane * 4);
    v4f* dst = (v4f*)(out + (size_t)S_ELEMS + (size_t)p * 128) + lane;
    __builtin_nontemporal_store(val, dst);
  }
}

// ---------------------------------------------------------------------------
// Kernel 3: collapse row (s_out row 0)
// ---------------------------------------------------------------------------
__global__ void collapse_kernel(const float* __restrict__ ct,
                                const float* __restrict__ cw,
                                float* __restrict__ out)
{
  const int c = threadIdx.x;
  if (c < S_DIM) out[c] = cw[0] * ct[c];
}

// ---------------------------------------------------------------------------
extern "C" void kernel_launch(void* const* d_in, const int* in_sizes, int n_in,
                              void* d_out, int out_size, void* d_ws, size_t ws_size,
                              hipStream_t stream)
{
  (void)in_sizes; (void)n_in; (void)out_size; (void)d_ws; (void)ws_size;
  // setup_inputs() dict order:
  const float* hd      = (const float*)d_in[0];
  const float* mhc     = (const float*)d_in[2];
  const int*   mhc_idx = (const int*)  d_in[3];
  const float* pep     = (const float*)d_in[4];
  const int*   pep_idx = (const int*)  d_in[5];
  const float* lv      = (const float*)d_in[6];
  const int*   lv_idx  = (const int*)  d_in[7];
  const float* lj      = (const float*)d_in[8];
  const int*   lj_idx  = (const int*)  d_in[9];
  const float* hv      = (const float*)d_in[10];
  const int*   hv_idx  = (const int*)  d_in[11];
  const float* hj      = (const float*)d_in[12];
  const int*   hj_idx  = (const int*)  d_in[13];
  const float* mask    = (const float*)d_in[14];
  const float* seq_W   = (const float*)d_in[15];
  const float* seq_b   = (const float*)d_in[16];
  const float* pos_W   = (const float*)d_in[17];
  const float* pos_b   = (const float*)d_in[18];
  const float* tab1    = (const float*)d_in[19];
  const float* b1      = (const float*)d_in[20];
  const float* tab2    = (const float*)d_in[21];
  const float* b2      = (const float*)d_in[22];
  const float* ct      = (const float*)d_in[23];
  const float* cw      = (const float*)d_in[24];
  const float* rw      = (const float*)d_in[25];
  float* out = (float*)d_out;

  collapse_kernel<<<1, S_DIM, 0, stream>>>(ct, cw, out);

  emb_kernel<<<TILES_TOT / WPB, 256, 0, stream>>>(
      hd, mask, mhc, pep, lv, lj, hv, hj,
      mhc_idx, pep_idx, lv_idx, lj_idx, hv_idx, hj_idx,
      seq_W, seq_b, pos_W, pos_b, rw, out);

  z_kernel<<<(ZPAIRS + WPB - 1) / WPB, 256, 0, stream>>>(tab1, b1, tab2, b2, out);
}